// GCNConv_55490977464723
// MI455X (gfx1250) — compile-verified
//
#include <hip/hip_runtime.h>

#define D 128
#define LDS_PITCH 132   // 128 + 4 dword pad: conflict-free b64 LDS reads of 128-strided rows

typedef __attribute__((ext_vector_type(2))) float v2f;
typedef __attribute__((ext_vector_type(8))) float v8f;

__global__ void zero_deg_kernel(int* __restrict__ deg, int n) {
    int i = blockIdx.x * blockDim.x + threadIdx.x;
    if (i < n) deg[i] = 0;
}

__global__ void count_deg_kernel(const int* __restrict__ row, int* __restrict__ deg, int E) {
    int e = blockIdx.x * blockDim.x + threadIdx.x;
    if (e < E) atomicAdd(&deg[row[e]], 1);
}

__global__ void dinv_kernel(const int* __restrict__ deg, float* __restrict__ dinv, int n) {
    int i = blockIdx.x * blockDim.x + threadIdx.x;
    if (i < n) {
        float d = (float)(deg[i] + 1);   // +1 for self-loop
        dinv[i] = rsqrtf(d);             // deg >= 1 always
    }
}

// y = x @ W^T using V_WMMA_F32_16X16X4_F32.
// 8 waves per block share one LDS-staged copy of W (64KB, padded pitch).
// Each wave computes a 16(M) x 128(N) strip: whole A strip preloaded to
// registers, B frags batched from LDS, 8 accumulators, 32 k-steps of 4.
__global__ void __launch_bounds__(256) gemm_xw_kernel(const float* __restrict__ x,
                                                      const float* __restrict__ W,
                                                      float* __restrict__ y, int nstrips) {
    __shared__ float ldsW[D * LDS_PITCH];

    const int tid = threadIdx.x;
    // Cooperative fill: 128x128 floats as 4096 float4s, 16 per thread.
#pragma unroll
    for (int i = 0; i < 16; ++i) {
        int idx = tid + i * 256;         // float4 index
        int r = idx >> 5;                // 32 float4 per row
        int c = (idx & 31) * 4;
        float4 v = *(const float4*)(W + (size_t)r * D + c);
        *(float4*)(&ldsW[r * LDS_PITCH + c]) = v;
    }
    __syncthreads();

    const int lane  = tid & 31;
    const int wave  = tid >> 5;
    const int strip = blockIdx.x * 8 + wave;
    if (strip >= nstrips) return;        // wave-uniform, after barrier: EXEC all-ones below

    const int lm    = lane & 15;         // row (A) / col (B,D) within tile
    const int khalf = (lane >> 4) << 1;  // lanes 0-15: k+0,k+1 ; lanes 16-31: k+2,k+3
    const float* xrow = x + (size_t)(strip * 16 + lm) * D;

    // Preload entire A strip: 32 fragments, all loads in flight before one wait.
    v2f a[32];
#pragma unroll
    for (int ks = 0; ks < 32; ++ks)
        a[ks] = *(const v2f*)(xrow + ks * 4 + khalf);

    v8f acc[8];
#pragma unroll
    for (int t = 0; t < 8; ++t) acc[t] = {};

#pragma unroll 4
    for (int ks = 0; ks < 32; ++ks) {
        const int k = ks * 4 + khalf;
        v2f bf[8];                       // batch all 8 B frags -> one dscnt wait per k-step
#pragma unroll
        for (int t = 0; t < 8; ++t)
            bf[t] = *(const v2f*)(&ldsW[(t * 16 + lm) * LDS_PITCH + k]);
#pragma unroll
        for (int t = 0; t < 8; ++t)
            acc[t] = __builtin_amdgcn_wmma_f32_16x16x4_f32(
                false, a[ks], false, bf[t], (short)0, acc[t], false, false);
    }

    // D layout: VGPR v holds M = v (lanes 0-15) or M = v+8 (lanes 16-31), N = lane&15
    const int mhi = (lane >> 4) * 8;
#pragma unroll
    for (int t = 0; t < 8; ++t) {
#pragma unroll
        for (int v = 0; v < 8; ++v) {
            y[(size_t)(strip * 16 + mhi + v) * D + t * 16 + lm] = acc[t][v];
        }
    }
}

// out[i] = b + dinv[i]^2 * y[i]   (bias + self-loop term, no atomics needed)
__global__ void init_out_kernel(const float* __restrict__ y, const float* __restrict__ dinv,
                                const float* __restrict__ bias, float* __restrict__ out, int n) {
    int gtid = blockIdx.x * blockDim.x + threadIdx.x;
    int i = gtid >> 5;
    int f = (gtid & 31) * 4;
    if (i >= n) return;
    float s = dinv[i];
    s = s * s;
    const float4 yv = *(const float4*)(y + (size_t)i * D + f);
    const float4 bv = *(const float4*)(bias + f);
    float4 o;
    o.x = bv.x + s * yv.x;
    o.y = bv.y + s * yv.y;
    o.z = bv.z + s * yv.z;
    o.w = bv.w + s * yv.w;
    *(float4*)(out + (size_t)i * D + f) = o;
}

// One wave per edge: out[col] += dinv[row]*dinv[col] * y[row]  (native f32 atomics)
__global__ void edge_scatter_kernel(const int* __restrict__ row, const int* __restrict__ col,
                                    const float* __restrict__ dinv, const float* __restrict__ y,
                                    float* __restrict__ out, int E) {
    int gtid = blockIdx.x * blockDim.x + threadIdx.x;
    int e    = gtid >> 5;
    int lane = gtid & 31;
    if (e >= E) return;
    int r = row[e], c = col[e];
    float norm = dinv[r] * dinv[c];
    const float4 v = *(const float4*)(y + (size_t)r * D + lane * 4);
    float* dst = out + (size_t)c * D + lane * 4;
    unsafeAtomicAdd(dst + 0, norm * v.x);
    unsafeAtomicAdd(dst + 1, norm * v.y);
    unsafeAtomicAdd(dst + 2, norm * v.z);
    unsafeAtomicAdd(dst + 3, norm * v.w);
}

extern "C" void kernel_launch(void* const* d_in, const int* in_sizes, int n_in,
                              void* d_out, int out_size, void* d_ws, size_t ws_size,
                              hipStream_t stream) {
    const float* x    = (const float*)d_in[0];
    const int*   ei   = (const int*)d_in[1];
    const float* W    = (const float*)d_in[2];
    const float* bias = (const float*)d_in[3];

    const int N = in_sizes[0] / D;
    const int E = in_sizes[1] / 2;
    const int* row = ei;
    const int* col = ei + E;

    char* ws    = (char*)d_ws;
    int*   deg  = (int*)ws;                          // N ints
    float* dinv = (float*)(ws + (size_t)N * 4);      // N floats
    float* y    = (float*)(ws + (size_t)N * 8);      // N*128 floats (16B aligned)
    float* out  = (float*)d_out;

    zero_deg_kernel<<<(N + 255) / 256, 256, 0, stream>>>(deg, N);
    count_deg_kernel<<<(E + 255) / 256, 256, 0, stream>>>(row, deg, E);
    dinv_kernel<<<(N + 255) / 256, 256, 0, stream>>>(deg, dinv, N);

    const int nstrips = (N + 15) / 16;               // N=100000 -> 6250 exact
    gemm_xw_kernel<<<(nstrips + 7) / 8, 256, 0, stream>>>(x, W, y, nstrips);

    init_out_kernel<<<(N * 32 + 255) / 256, 256, 0, stream>>>(y, dinv, bias, out, N);
    edge_scatter_kernel<<<(E * 32 + 255) / 256, 256, 0, stream>>>(row, col, dinv, y, out, E);
}